// EnergyStatistics_661424963713
// MI455X (gfx1250) — compile-verified
//
#include <hip/hip_runtime.h>
#include <math.h>

typedef __attribute__((ext_vector_type(2))) float v2f;
typedef __attribute__((ext_vector_type(4))) float v4f;
typedef __attribute__((ext_vector_type(8))) float v8f;

namespace {
constexpr int B      = 32768;
constexpr int D      = 128;
constexpr int NC     = 32;
constexpr int K      = 100;
constexpr int KP     = 112;                 // K padded to 7 WMMA tiles of 16
constexpr int NT     = KP / 16;             // 7
constexpr int NTILES = NT * (NT + 1) / 2;   // 28 upper-triangular tile pairs
constexpr int SLICES = 16;                  // B-slices for occupancy
constexpr float BIGF = 1e30f;

// workspace layout (float offsets)
constexpr int OFF_COUNTS = 0;                      // [NC][K]
constexpr int OFF_PERSUM = OFF_COUNTS + NC * K;    // [NC][K]
constexpr int OFF_SUMS   = OFF_PERSUM + NC * K;    // [NC][K][D]
constexpr int OFF_CENT   = OFF_SUMS + NC * K * D;  // [NC][KP][D]
constexpr int OFF_CN     = OFF_CENT + NC * KP * D; // [NC][KP]
constexpr int OFF_ENERGY = OFF_CN + NC * KP;       // [NC][4]
constexpr int ZERO_N     = OFF_SUMS + NC * K * D;  // counts + per_sum + sums
} // namespace

__global__ void k_zero(float* __restrict__ p, int n) {
    for (int i = blockIdx.x * blockDim.x + threadIdx.x; i < n;
         i += gridDim.x * blockDim.x)
        p[i] = 0.0f;
}

// ---- Pass 1: segment counts + feature sums (LDS accumulate, global merge) ----
__global__ void k_accum(const float* __restrict__ feat, const int* __restrict__ assign,
                        float* __restrict__ g_counts, float* __restrict__ g_sums) {
    __shared__ float s_sums[K * D];   // 50 KB
    __shared__ float s_cnt[K];
    const int c   = blockIdx.x;       // config
    const int tid = threadIdx.x;
    for (int i = tid; i < K * D; i += 256) s_sums[i] = 0.0f;
    if (tid < K) s_cnt[tid] = 0.0f;
    __syncthreads();

    const int per = B / SLICES;
    const int b0  = blockIdx.y * per;
    for (int b = b0 + tid; b < b0 + per; b += 256) {
        const int k = assign[b * NC + c];
        atomicAdd(&s_cnt[k], 1.0f);
        const v4f* f4 = (const v4f*)(feat + (size_t)b * D);
        float* dst = &s_sums[k * D];
#pragma unroll 8
        for (int d4 = 0; d4 < D / 4; ++d4) {
            v4f v = f4[d4];
            atomicAdd(&dst[4 * d4 + 0], v.x);
            atomicAdd(&dst[4 * d4 + 1], v.y);
            atomicAdd(&dst[4 * d4 + 2], v.z);
            atomicAdd(&dst[4 * d4 + 3], v.w);
        }
    }
    __syncthreads();
    float* gs = g_sums + (size_t)c * K * D;
    for (int i = tid; i < K * D; i += 256) atomicAdd(&gs[i], s_sums[i]);
    if (tid < K) atomicAdd(&g_counts[c * K + tid], s_cnt[tid]);
}

// ---- Pass 2: centroids (padded to KP rows) + squared norms ----
__global__ void k_centroid(const float* __restrict__ g_counts,
                           const float* __restrict__ g_sums,
                           float* __restrict__ g_cent, float* __restrict__ g_cn) {
    const int c  = blockIdx.x / KP;
    const int kp = blockIdx.x % KP;
    const int d  = threadIdx.x;   // 128 threads
    float v = 0.0f;
    if (kp < K) {
        const float cnt = g_counts[c * K + kp];
        v = g_sums[(size_t)c * K * D + kp * D + d] / fmaxf(cnt, 1.0f);
    }
    g_cent[(size_t)c * KP * D + kp * D + d] = v;
    __shared__ float red[128];
    red[d] = v * v;
    __syncthreads();
    for (int s = 64; s > 0; s >>= 1) {
        if (d < s) red[d] += red[d + s];
        __syncthreads();
    }
    if (d == 0) g_cn[c * KP + kp] = red[0];
}

// ---- Pass 3: per-sample distance to own centroid, per-cluster sum ----
__global__ void k_dist(const float* __restrict__ feat, const int* __restrict__ assign,
                       const float* __restrict__ g_cent, float* __restrict__ g_persum) {
    __shared__ float s_cent[K * D];   // 50 KB
    __shared__ float s_per[K];
    const int c   = blockIdx.x;
    const int tid = threadIdx.x;
    const float* cent_c = g_cent + (size_t)c * KP * D;  // rows 0..K-1 contiguous
    for (int i = tid; i < K * D; i += 256) s_cent[i] = cent_c[i];
    if (tid < K) s_per[tid] = 0.0f;
    __syncthreads();

    const int per = B / SLICES;
    const int b0  = blockIdx.y * per;
    for (int b = b0 + tid; b < b0 + per; b += 256) {
        const int k = assign[b * NC + c];
        const v4f* f4 = (const v4f*)(feat + (size_t)b * D);
        const v4f* c4 = (const v4f*)(&s_cent[k * D]);
        float acc = 0.0f;
#pragma unroll 8
        for (int d4 = 0; d4 < D / 4; ++d4) {
            v4f f = f4[d4], cc = c4[d4];
            const float dx = f.x - cc.x, dy = f.y - cc.y;
            const float dz = f.z - cc.z, dw = f.w - cc.w;
            acc = fmaf(dx, dx, acc); acc = fmaf(dy, dy, acc);
            acc = fmaf(dz, dz, acc); acc = fmaf(dw, dw, acc);
        }
        atomicAdd(&s_per[k], sqrtf(fmaxf(acc, 1e-12f)));
    }
    __syncthreads();
    if (tid < K) atomicAdd(&g_persum[c * K + tid], s_per[tid]);
}

// ---- Pass 4: per-config stats; inter-centroid Gram via V_WMMA_F32_16X16X4_F32 ----
__global__ void k_stats(const float* __restrict__ g_counts, const float* __restrict__ g_persum,
                        const float* __restrict__ g_cent, const float* __restrict__ g_cn,
                        const float* __restrict__ rmean, const float* __restrict__ rvar,
                        float* __restrict__ energy) {
    __shared__ float s_cent[KP * D];  // 56 KB
    __shared__ float s_cnt[K];
    __shared__ float s_per[K];
    __shared__ float s_cn[KP];
    __shared__ float s_total, s_plogp, s_nonempty, s_nmulti, s_hasum, s_suminter, s_npair;
    __shared__ int s_minbits, s_maxbits;

    const int c   = blockIdx.x;
    const int tid = threadIdx.x;
    const float* cent_c = g_cent + (size_t)c * KP * D;
    for (int i = tid; i < KP * D; i += 256) s_cent[i] = cent_c[i];
    if (tid < K) { s_cnt[tid] = g_counts[c * K + tid]; s_per[tid] = g_persum[c * K + tid]; }
    if (tid < KP) s_cn[tid] = g_cn[c * KP + tid];
    if (tid == 0) {
        s_total = 0.0f; s_plogp = 0.0f; s_nonempty = 0.0f; s_nmulti = 0.0f;
        s_hasum = 0.0f; s_suminter = 0.0f; s_npair = 0.0f;
        s_minbits = __float_as_int(BIGF);
        s_maxbits = 0;                   // inter >= sqrt(1e-12) > 0, safe floor
    }
    __syncthreads();

    if (tid < K) {
        const float cnt = s_cnt[tid];
        atomicAdd(&s_total, cnt);
        if (cnt > 0.0f) atomicAdd(&s_nonempty, 1.0f);
        if (cnt > 1.0f) {
            atomicAdd(&s_nmulti, 1.0f);
            const float pm = s_per[tid] / fmaxf(cnt, 1.0f);
            atomicAdd(&s_hasum, pm);
            atomicMin(&s_minbits, __float_as_int(pm));   // non-negative: bit order == float order
        }
    }
    __syncthreads();
    if (tid < K) {
        const float p = s_cnt[tid] / s_total + 1e-10f;
        atomicAdd(&s_plogp, p * logf(p));
    }

    // Gram = C (KPxD) * C^T via f32 WMMA; 8 waves cover 28 upper-tri tiles.
    const int lane = tid & 31;
    const int wave = tid >> 5;
    const int row  = lane & 15;
    const int hi   = lane >> 4;   // lanes 16-31 hold K+2/K+3 (A) / upper half rows (D)
    for (int t = wave; t < NTILES; t += 8) {       // wave-uniform -> EXEC stays full
        int ti = 0, rem = t;
        while (rem >= NT - ti) { rem -= NT - ti; ++ti; }
        const int tj = ti + rem;
        const float* arow = &s_cent[(ti * 16 + row) * D];
        const float* brow = &s_cent[(tj * 16 + row) * D];
        v8f acc = {};
#pragma unroll 8
        for (int kk = 0; kk < D; kk += 4) {
            v2f a, b;
            a.x = arow[kk + 2 * hi + 0];
            a.y = arow[kk + 2 * hi + 1];
            b.x = brow[kk + 2 * hi + 0];
            b.y = brow[kk + 2 * hi + 1];
            acc = __builtin_amdgcn_wmma_f32_16x16x4_f32(
                /*neg_a=*/false, a, /*neg_b=*/false, b,
                /*c_mod=*/(short)0, acc, /*reuse_a=*/false, /*reuse_b=*/false);
        }
#pragma unroll
        for (int r = 0; r < 8; ++r) {
            const int m = ti * 16 + r + 8 * hi;    // D layout: VGPR r -> M=r (+8 upper half)
            const int n = tj * 16 + row;
            const float d2    = s_cn[m] + s_cn[n] - 2.0f * acc[r];
            const float inter = sqrtf(fmaxf(d2, 1e-12f));
            const bool pair = (m < n) && (m < K) && (n < K) &&
                              (s_cnt[m] > 0.0f) && (s_cnt[n] > 0.0f);
            if (pair) {
                atomicAdd(&s_suminter, inter);
                atomicAdd(&s_npair, 1.0f);
                atomicMax(&s_maxbits, __float_as_int(inter));
            }
        }
    }
    __syncthreads();

    if (tid == 0) {
        const float H        = -s_plogp;
        const bool many      = s_nonempty > 1.0f;
        const bool has_multi = s_nmulti > 0.0f;
        const bool has_pair  = s_npair > 0.0f;
        float h_a = has_multi ? s_hasum / fmaxf(s_nmulti, 1.0f) : 0.0f;
        const float min_intra = has_multi ? __int_as_float(s_minbits) : 0.0f;
        float h_r = has_pair ? s_suminter / fmaxf(s_npair, 1.0f) : 0.0f;
        const float max_inter = has_pair ? __int_as_float(s_maxbits) : 0.0f;
        float delta = max_inter - min_intra;
        if (!many) { h_a = 0.0f; h_r = 0.0f; delta = 0.0f; }
        const float e[4] = {H, h_a, h_r, delta};
#pragma unroll
        for (int j = 0; j < 4; ++j) {
            energy[c * 4 + j] =
                (e[j] - rmean[c * 4 + j]) / (sqrtf(rvar[c * 4 + j]) + 1e-8f);
        }
    }
}

// ---- Pass 5: broadcast [NC][4] table over B rows, streaming NT stores ----
__global__ void k_broadcast(const float* __restrict__ energy, v4f* __restrict__ out4) {
    __shared__ v4f tbl[NC];
    if (threadIdx.x < NC) tbl[threadIdx.x] = ((const v4f*)energy)[threadIdx.x];
    __syncthreads();
    const int stride = gridDim.x * blockDim.x;    // multiple of 32 -> c is loop-invariant
    int i = blockIdx.x * blockDim.x + threadIdx.x;
    const v4f v = tbl[i & (NC - 1)];
    for (; i < B * NC; i += stride)
        __builtin_nontemporal_store(v, &out4[i]);
}

extern "C" void kernel_launch(void* const* d_in, const int* in_sizes, int n_in,
                              void* d_out, int out_size, void* d_ws, size_t ws_size,
                              hipStream_t stream) {
    (void)in_sizes; (void)n_in; (void)out_size; (void)ws_size;
    const float* features = (const float*)d_in[0];        // [B, D]  f32
    const int*   assign   = (const int*)d_in[1];          // [B, NC] i32
    const float* rmean    = (const float*)d_in[2];        // [NC, 4] f32
    const float* rvar     = (const float*)d_in[3];        // [NC, 4] f32
    float* out = (float*)d_out;                           // [B, NC, 4] f32
    float* ws  = (float*)d_ws;

    float* w_counts = ws + OFF_COUNTS;
    float* w_persum = ws + OFF_PERSUM;
    float* w_sums   = ws + OFF_SUMS;
    float* w_cent   = ws + OFF_CENT;
    float* w_cn     = ws + OFF_CN;
    float* w_energy = ws + OFF_ENERGY;

    k_zero<<<1024, 256, 0, stream>>>(ws, ZERO_N);
    k_accum<<<dim3(NC, SLICES), 256, 0, stream>>>(features, assign, w_counts, w_sums);
    k_centroid<<<NC * KP, 128, 0, stream>>>(w_counts, w_sums, w_cent, w_cn);
    k_dist<<<dim3(NC, SLICES), 256, 0, stream>>>(features, assign, w_cent, w_persum);
    k_stats<<<NC, 256, 0, stream>>>(w_counts, w_persum, w_cent, w_cn, rmean, rvar, w_energy);
    k_broadcast<<<2048, 256, 0, stream>>>(w_energy, (v4f*)out);
}